// Autoregression_54374285967486
// MI455X (gfx1250) — compile-verified
//
#include <hip/hip_runtime.h>
#include <hip/hip_bf16.h>
#include <math.h>

typedef __attribute__((ext_vector_type(2))) float v2f;
typedef __attribute__((ext_vector_type(8))) float v8f;

#define ROWS_PER_WG 128
#define NJ 23
#define NTOT 3014656L /* 131072 * 23 */

// One Jacobi (Givens) rotation zeroing B[p][q] of the symmetric 3x3; (apr,aqr)
// are the off terms coupling to the third index; V accumulates right vectors.
__device__ __forceinline__ void jrot(float& app, float& aqq, float& apq,
                                     float& apr, float& aqr,
                                     float* V, int p, int q) {
  float a = apq;
  if (fabsf(a) < 1e-30f) return;
  float tau = (aqq - app) / (2.0f * a);
  float t = copysignf(1.0f, tau) / (fabsf(tau) + sqrtf(1.0f + tau * tau));
  float c = 1.0f / sqrtf(1.0f + t * t);
  float s = t * c;
  app -= t * a;
  aqq += t * a;
  apq = 0.0f;
  float pr = apr, qr = aqr;
  apr = c * pr - s * qr;
  aqr = s * pr + c * qr;
#pragma unroll
  for (int i = 0; i < 3; ++i) {
    float vp = V[i * 3 + p], vq = V[i * 3 + q];
    V[i * 3 + p] = c * vp - s * vq;
    V[i * 3 + q] = s * vp + c * vq;
  }
}

__global__ __launch_bounds__(256) void smpl_mlp_svd_kernel(
    const float* __restrict__ feature,
    const float* __restrict__ w0, const float* __restrict__ b0,
    const float* __restrict__ w1, const float* __restrict__ b1,
    const float* __restrict__ w2, const float* __restrict__ b2,
    float* __restrict__ out) {
  // B-operand weights packed as float2 {w[2kp][n], w[2kp+1][n]} in [kpair][n]
  // layout: one ds_load_b64 per WMMA B fragment. Pair-row strides are 16 mod 32
  // so the two half-waves (pair-rows 2kp and 2kp+1) hit disjoint bank groups.
  __shared__ v2f sW0p[36][144];   // w0aug^T pairs: k=0..71 (rows 0..2 zero), n=0..127
  __shared__ v2f sW1p[64][144];   // w1^T pairs
  __shared__ v2f sW2p[64][80];    // w2aug^T pairs: n=0..79, n>=69 zero
  __shared__ float sH[128][132];  // hidden activations, row-major per batch row
  __shared__ float sR[128][72];   // layer2 output (rvec values), per batch row
  __shared__ float sB0[128], sB1[128], sB2[80];

  const int tid = threadIdx.x;

  // ---- Stage weights into LDS (once per workgroup), pre-paired for b64 ----
  for (int idx = tid; idx < 36 * 128; idx += 256) {
    int kp = idx >> 7, n = idx & 127;
    int k0 = 2 * kp, k1 = 2 * kp + 1;
    v2f t;
    t.x = (k0 < 3) ? 0.0f : w0[n * 69 + (k0 - 3)];
    t.y = (k1 < 3) ? 0.0f : w0[n * 69 + (k1 - 3)];
    sW0p[kp][n] = t;
  }
  for (int idx = tid; idx < 64 * 128; idx += 256) {
    int kp = idx >> 7, n = idx & 127;
    v2f t;
    t.x = w1[n * 128 + 2 * kp];
    t.y = w1[n * 128 + 2 * kp + 1];
    sW1p[kp][n] = t;
  }
  for (int idx = tid; idx < 64 * 80; idx += 256) {
    int kp = idx / 80, n = idx % 80;
    v2f t;
    t.x = (n < 69) ? w2[n * 128 + 2 * kp] : 0.0f;
    t.y = (n < 69) ? w2[n * 128 + 2 * kp + 1] : 0.0f;
    sW2p[kp][n] = t;
  }
  if (tid < 128) { sB0[tid] = b0[tid]; sB1[tid] = b1[tid]; }
  if (tid < 80) sB2[tid] = (tid < 69) ? b2[tid] : 0.0f;
  __syncthreads();

  // ---- WMMA fragment coordinates (wave32; 16x16x4 f32 layouts per ISA) ----
  const int lane = tid & 31;
  const int wv = tid >> 5;          // wave id: owns rows [wv*16, wv*16+16)
  const int half = lane >> 4;       // 0: lanes 0-15, 1: lanes 16-31
  const int nlo = lane & 15;
  const int off = half * 2;         // A/B: K offset +0 or +2 per half-wave
  const int rowL = wv * 16 + nlo;   // local M row this lane loads for A
  const long grow = (long)blockIdx.x * ROWS_PER_WG + rowL;
  const float* featRow = feature + grow * 72;

  // ---- Layer 0: feature(16x72) @ w0aug^T(72x128) ----
  v8f acc0[8] = {};
  for (int kp = 0; kp < 18; ++kp) {
    int kbase = 4 * kp;
    v2f a;
    __builtin_memcpy(&a, featRow + kbase + off, 8);  // global_load_b64
    int bp = 2 * kp + half;
#pragma unroll
    for (int t = 0; t < 8; ++t) {
      v2f b = sW0p[bp][nlo + 16 * t];                // ds_load_b64
      acc0[t] = __builtin_amdgcn_wmma_f32_16x16x4_f32(
          false, a, false, b, (short)0, acc0[t], false, false);
    }
  }
#pragma unroll
  for (int t = 0; t < 8; ++t) {
    int n = nlo + 16 * t;
#pragma unroll
    for (int e = 0; e < 8; ++e) {
      int m = e + 8 * half;
      float v = acc0[t][e] + sB0[n];
      sH[wv * 16 + m][n] = v > 0.0f ? v : 0.0f;
    }
  }
  __syncthreads();

  // ---- Layer 1: h(16x128) @ w1^T(128x128) ----
  v8f acc1[8] = {};
  for (int kp = 0; kp < 32; ++kp) {
    int kbase = 4 * kp;
    v2f a;
    __builtin_memcpy(&a, &sH[rowL][kbase + off], 8); // ds_load_b64
    int bp = 2 * kp + half;
#pragma unroll
    for (int t = 0; t < 8; ++t) {
      v2f b = sW1p[bp][nlo + 16 * t];
      acc1[t] = __builtin_amdgcn_wmma_f32_16x16x4_f32(
          false, a, false, b, (short)0, acc1[t], false, false);
    }
  }
  __syncthreads();   // all reads of sH done before overwriting it
#pragma unroll
  for (int t = 0; t < 8; ++t) {
    int n = nlo + 16 * t;
#pragma unroll
    for (int e = 0; e < 8; ++e) {
      int m = e + 8 * half;
      float v = acc1[t][e] + sB1[n];
      sH[wv * 16 + m][n] = v > 0.0f ? v : 0.0f;
    }
  }
  __syncthreads();

  // ---- Layer 2: h(16x128) @ w2aug^T(128x80) ----
  v8f acc2[5] = {};
  for (int kp = 0; kp < 32; ++kp) {
    int kbase = 4 * kp;
    v2f a;
    __builtin_memcpy(&a, &sH[rowL][kbase + off], 8);
    int bp = 2 * kp + half;
#pragma unroll
    for (int t = 0; t < 5; ++t) {
      v2f b = sW2p[bp][nlo + 16 * t];
      acc2[t] = __builtin_amdgcn_wmma_f32_16x16x4_f32(
          false, a, false, b, (short)0, acc2[t], false, false);
    }
  }
#pragma unroll
  for (int t = 0; t < 5; ++t) {
    int n = nlo + 16 * t;
    if (n < 72) {
#pragma unroll
      for (int e = 0; e < 8; ++e) {
        int m = e + 8 * half;
        sR[wv * 16 + m][n] = acc2[t][e] + sB2[n];
      }
    }
  }
  __syncthreads();

  // ---- Per-joint: Rodrigues -> 3x3 SVD (Jacobi on F^T F) -> rotmat ----
  // All results are write-once streaming (never re-read): nontemporal stores
  // keep 470MB of output from churning L2 (weights stay resident for all WGs).
  float* oF = out;
  float* oU = out + 9L * NTOT;
  float* oS = out + 18L * NTOT;
  float* oV = out + 21L * NTOT;
  float* oR = out + 30L * NTOT;

  for (int jj = tid; jj < ROWS_PER_WG * NJ; jj += 256) {
    int lr = jj / NJ;
    int j = jj % NJ;
    long g = ((long)blockIdx.x * ROWS_PER_WG + lr) * NJ + j;

    float r0 = sR[lr][3 * j + 0];
    float r1 = sR[lr][3 * j + 1];
    float r2 = sR[lr][3 * j + 2];

    // Rodrigues (matches reference exactly, incl. the 1e-5 in theta)
    float theta = sqrtf(1e-5f + r0 * r0 + r1 * r1 + r2 * r2);
    float it = 1.0f / theta;
    float x = r0 * it, y = r1 * it, z = r2 * it;
    float sth, cth;
    sincosf(theta, &sth, &cth);
    float omc = 1.0f - cth;
    float F[9];
    F[0] = x * x + (1.0f - x * x) * cth;
    F[1] = x * y * omc - z * sth;
    F[2] = x * z * omc + y * sth;
    F[3] = x * y * omc + z * sth;
    F[4] = y * y + (1.0f - y * y) * cth;
    F[5] = y * z * omc - x * sth;
    F[6] = x * z * omc - y * sth;
    F[7] = y * z * omc + x * sth;
    F[8] = z * z + (1.0f - z * z) * cth;

    // Bsym = F^T F
    float b00 = F[0] * F[0] + F[3] * F[3] + F[6] * F[6];
    float b11 = F[1] * F[1] + F[4] * F[4] + F[7] * F[7];
    float b22 = F[2] * F[2] + F[5] * F[5] + F[8] * F[8];
    float b01 = F[0] * F[1] + F[3] * F[4] + F[6] * F[7];
    float b02 = F[0] * F[2] + F[3] * F[5] + F[6] * F[8];
    float b12 = F[1] * F[2] + F[4] * F[5] + F[7] * F[8];

    float V[9] = {1.f, 0.f, 0.f, 0.f, 1.f, 0.f, 0.f, 0.f, 1.f};
#pragma unroll
    for (int sw = 0; sw < 4; ++sw) {
      jrot(b00, b11, b01, b02, b12, V, 0, 1);
      jrot(b00, b22, b02, b01, b12, V, 0, 2);
      jrot(b11, b22, b12, b01, b02, V, 1, 2);
    }

    // Sort eigenvalues descending, permuting V columns
    float d0 = b00, d1 = b11, d2 = b22;
    if (d0 < d1) {
      float tmp = d0; d0 = d1; d1 = tmp;
#pragma unroll
      for (int i = 0; i < 3; ++i) { float v = V[i*3]; V[i*3] = V[i*3+1]; V[i*3+1] = v; }
    }
    if (d0 < d2) {
      float tmp = d0; d0 = d2; d2 = tmp;
#pragma unroll
      for (int i = 0; i < 3; ++i) { float v = V[i*3]; V[i*3] = V[i*3+2]; V[i*3+2] = v; }
    }
    if (d1 < d2) {
      float tmp = d1; d1 = d2; d2 = tmp;
#pragma unroll
      for (int i = 0; i < 3; ++i) { float v = V[i*3+1]; V[i*3+1] = V[i*3+2]; V[i*3+2] = v; }
    }

    float s0 = sqrtf(fmaxf(d0, 0.0f));
    float s1 = sqrtf(fmaxf(d1, 0.0f));
    float s2 = sqrtf(fmaxf(d2, 0.0f));
    float i0 = 1.0f / fmaxf(s0, 1e-30f);
    float i1 = 1.0f / fmaxf(s1, 1e-30f);
    float i2 = 1.0f / fmaxf(s2, 1e-30f);

    // U[:,k] = F * V[:,k] / s_k
    float U[9];
#pragma unroll
    for (int i = 0; i < 3; ++i) {
      float f0 = F[i * 3 + 0], f1 = F[i * 3 + 1], f2 = F[i * 3 + 2];
      U[i * 3 + 0] = (f0 * V[0] + f1 * V[3] + f2 * V[6]) * i0;
      U[i * 3 + 1] = (f0 * V[1] + f1 * V[4] + f2 * V[7]) * i1;
      U[i * 3 + 2] = (f0 * V[2] + f1 * V[5] + f2 * V[8]) * i2;
    }

    float detU = U[0] * (U[4] * U[8] - U[5] * U[7])
               - U[1] * (U[3] * U[8] - U[5] * U[6])
               + U[2] * (U[3] * U[7] - U[4] * U[6]);
    float detV = V[0] * (V[4] * V[8] - V[5] * V[7])
               - V[1] * (V[3] * V[8] - V[5] * V[6])
               + V[2] * (V[3] * V[7] - V[4] * V[6]);
    float dd = detU * detV;

    long b9 = g * 9;
    // rotmat = U' V'^T with third columns scaled by detU / detV (folded as dd)
#pragma unroll
    for (int i = 0; i < 3; ++i) {
#pragma unroll
      for (int jc = 0; jc < 3; ++jc) {
        float r = U[i * 3 + 0] * V[jc * 3 + 0] +
                  U[i * 3 + 1] * V[jc * 3 + 1] +
                  dd * U[i * 3 + 2] * V[jc * 3 + 2];
        __builtin_nontemporal_store(r, oR + b9 + i * 3 + jc);
      }
    }
#pragma unroll
    for (int i = 0; i < 9; ++i) {
      __builtin_nontemporal_store(F[i], oF + b9 + i);
      __builtin_nontemporal_store(U[i], oU + b9 + i);
      __builtin_nontemporal_store(V[i], oV + b9 + i);
    }
    __builtin_nontemporal_store(s0, oS + g * 3 + 0);
    __builtin_nontemporal_store(s1, oS + g * 3 + 1);
    __builtin_nontemporal_store(s2, oS + g * 3 + 2);
  }
}

extern "C" void kernel_launch(void* const* d_in, const int* in_sizes, int n_in,
                              void* d_out, int out_size, void* d_ws, size_t ws_size,
                              hipStream_t stream) {
  (void)in_sizes; (void)n_in; (void)out_size; (void)d_ws; (void)ws_size;
  const float* feature = (const float*)d_in[0];
  const float* w0 = (const float*)d_in[1];
  const float* b0 = (const float*)d_in[2];
  const float* w1 = (const float*)d_in[3];
  const float* b1 = (const float*)d_in[4];
  const float* w2 = (const float*)d_in[5];
  const float* b2 = (const float*)d_in[6];
  float* out = (float*)d_out;

  dim3 grid(131072 / ROWS_PER_WG);  // 1024 workgroups x 128 rows
  dim3 block(256);                  // 8 wave32 waves
  hipLaunchKernelGGL(smpl_mlp_svd_kernel, grid, block, 0, stream,
                     feature, w0, b0, w1, b1, w2, b2, out);
}